// GCN_layer_53807350284437
// MI455X (gfx1250) — compile-verified
//
#include <hip/hip_runtime.h>
#include <math.h>

typedef __attribute__((ext_vector_type(2))) float v2f;
typedef __attribute__((ext_vector_type(8))) float v8f;

#define NGRAPHS 64

// ---------- helpers ----------
__device__ __forceinline__ float sel_load(const float* __restrict__ p, int idx, bool valid) {
  // branch-free guarded load: clamp index, cndmask the value (keeps EXEC full for WMMA)
  int i = valid ? idx : 0;
  float v = p[i];
  return valid ? v : 0.0f;
}

// A-fragment load with optional fused bias+ReLU (previous layer's epilogue)
template<bool PRE>
__device__ __forceinline__ float ld_a(const float* __restrict__ row,
                                      const float* __restrict__ bias,
                                      int f, bool valid) {
  int i = valid ? f : 0;
  float v = row[i];
  if (PRE) v = fmaxf(v + bias[i], 0.0f);
  return valid ? v : 0.0f;
}

// ---------- degree / normalization ----------
__global__ void k_deg_init(float* __restrict__ deg, int n) {
  int i = blockIdx.x * blockDim.x + threadIdx.x;
  if (i < n) deg[i] = 1.0f;  // self-loop contributes 1
}

__global__ void k_deg_accum(const int* __restrict__ dst, float* __restrict__ deg, int ne) {
  int e = blockIdx.x * blockDim.x + threadIdx.x;
  if (e < ne) unsafeAtomicAdd(&deg[dst[e]], 1.0f);   // global_atomic_add_f32
}

__global__ void k_deg_rsqrt(float* __restrict__ deg, int n) {
  int i = blockIdx.x * blockDim.x + threadIdx.x;
  if (i < n) deg[i] = rsqrtf(deg[i]);  // deg >= 1 always (self-loops)
}

// ---------- WMMA GEMM + fused pre-act + fused self-loop epilogue ----------
// H[n,FOUT] = act(X) @ W ;  AGG[n,FOUT] = H * dinv^2   (self-loop init)
// act(X) = PRE ? relu(X + preb) : X.
// One wave owns one 16-col tile and grid-strides over 16-node tiles; B fragments
// (weights) are preloaded once per wave.
// VGPR layouts per ISA: A 16x4 f32 (lane-half -> K pairs {0,1}/{2,3});
// B 4x16 assumed analogous (VGPR r -> rows r, r+2); C/D: VGPR v -> rows v, v+8.
template<int FIN, int FOUT, bool PRE>
__global__ void __launch_bounds__(256) k_gemm_wmma(
    const float* __restrict__ X, const float* __restrict__ W,
    const float* __restrict__ preb, const float* __restrict__ dinv,
    float* __restrict__ H, float* __restrict__ AGG, int n, int ntStride) {
  constexpr int KPAD = (FIN + 3) & ~3;
  constexpr int NK   = KPAD / 4;
  constexpr int CT   = (FOUT + 15) / 16;
  int wave  = (blockIdx.x * blockDim.x + threadIdx.x) >> 5;   // wave-uniform
  int lane  = threadIdx.x & 31;
  int ctile = wave % CT;
  int nt0   = wave / CT;
  int half  = lane >> 4;          // 0 or 1
  int m     = lane & 15;
  int col   = ctile * 16 + m;
  bool cvld = (FOUT % 16 == 0) ? true : (col < FOUT);

  // preload all B fragments for this column tile (weights reused across node tiles)
  v2f bf[NK];
#pragma unroll
  for (int kk = 0; kk < NK; ++kk) {
    int ka = kk * 4 + 2 * half;
    bf[kk][0] = sel_load(W, ka * FOUT + col,       cvld && (ka     < FIN));
    bf[kk][1] = sel_load(W, (ka + 1) * FOUT + col, cvld && (ka + 1 < FIN));
  }

  int ntiles = (n + 15) >> 4;
  for (int nt = nt0; nt < ntiles; nt += ntStride) {
    int row0  = nt << 4;
    bool full = (row0 + 16 <= n);          // wave-uniform
    int node  = row0 + m;
    bool nvld = node < n;
    const float* Xr = X + (size_t)(nvld ? node : (n - 1)) * FIN;

    v8f c = {};
#pragma unroll
    for (int kk = 0; kk < NK; ++kk) {
      int ka = kk * 4 + 2 * half;
      v2f a;
      a[0] = ld_a<PRE>(Xr, preb, ka,     nvld && (ka     < FIN));
      a[1] = ld_a<PRE>(Xr, preb, ka + 1, nvld && (ka + 1 < FIN));
      c = __builtin_amdgcn_wmma_f32_16x16x4_f32(false, a, false, bf[kk],
                                                (short)0, c, false, false);
    }

    if (full) {                            // wave-uniform fast path: plain stores
#pragma unroll
      for (int v = 0; v < 8; ++v) {
        int row = row0 + v + 8 * half;
        if ((FOUT % 16) == 0 || cvld) {
          float hv = c[v];
          float di = dinv[row];            // broadcast within half-wave
          size_t o = (size_t)row * FOUT + col;
          H[o]   = hv;
          AGG[o] = hv * di * di;
        }
      }
    } else {                               // partial tail tile
#pragma unroll
      for (int v = 0; v < 8; ++v) {
        int row = row0 + v + 8 * half;
        if (row < n && cvld) {
          float hv = c[v];
          float di = dinv[row];
          size_t o = (size_t)row * FOUT + col;
          H[o]   = hv;
          AGG[o] = hv * di * di;
        }
      }
    }
  }
}

// ---------- edge scatter: agg[dst] += h[src] * dinv[src]*dinv[dst] ----------
template<int F>
__global__ void k_scatter(const int* __restrict__ src, const int* __restrict__ dst,
                          const float* __restrict__ dinv, const float* __restrict__ h,
                          float* __restrict__ agg, int ne) {
  constexpr int CPF = (F + 3) / 4;   // 4-feature chunks per edge
  long t = (long)blockIdx.x * blockDim.x + threadIdx.x;
  if (t >= (long)ne * CPF) return;
  int e  = (int)(t / CPF);
  int ch = (int)(t % CPF);
  int s = src[e], d = dst[e];
  float w = dinv[s] * dinv[d];
  int f0 = ch * 4;
  const float* hs = h   + (size_t)s * F + f0;
  float*       ad = agg + (size_t)d * F + f0;
  if constexpr ((F % 4) == 0) {
    float4 v = *reinterpret_cast<const float4*>(hs);   // 16B aligned (F=16/64)
    unsafeAtomicAdd(ad + 0, v.x * w);
    unsafeAtomicAdd(ad + 1, v.y * w);
    unsafeAtomicAdd(ad + 2, v.z * w);
    unsafeAtomicAdd(ad + 3, v.w * w);
  } else {
    int rem = F - f0; if (rem > 4) rem = 4;
    for (int f = 0; f < rem; ++f) unsafeAtomicAdd(ad + f, hs[f] * w);
  }
}

// ---------- pooling ----------
__global__ void k_zero(float* __restrict__ p, int n) {
  int i = blockIdx.x * blockDim.x + threadIdx.x;
  if (i < n) p[i] = 0.0f;
}

__global__ void k_pool(const float* __restrict__ h, const int* __restrict__ batch,
                       float* __restrict__ sums, float* __restrict__ cnt, int n) {
  int i = blockIdx.x * blockDim.x + threadIdx.x;
  if (i >= n) return;
  int g = batch[i];
#pragma unroll
  for (int f = 0; f < 6; ++f) unsafeAtomicAdd(&sums[g * 6 + f], h[(size_t)i * 6 + f]);
  unsafeAtomicAdd(&cnt[g], 1.0f);
}

// mean pool (+ layer-4 bias, which commutes with the mean) + log_softmax
__global__ void k_finalize(const float* __restrict__ sums, const float* __restrict__ cnt,
                           const float* __restrict__ b4, float* __restrict__ out) {
  int g = threadIdx.x;
  if (g >= NGRAPHS) return;
  float c = fmaxf(cnt[g], 1.0f);
  float v[6], mx = -INFINITY;
#pragma unroll
  for (int f = 0; f < 6; ++f) { v[f] = sums[g * 6 + f] / c + b4[f]; mx = fmaxf(mx, v[f]); }
  float s = 0.0f;
#pragma unroll
  for (int f = 0; f < 6; ++f) s += expf(v[f] - mx);
  float lse = mx + logf(s);
#pragma unroll
  for (int f = 0; f < 6; ++f) out[g * 6 + f] = v[f] - lse;
}

// ---------- host-side layer driver ----------
template<int FIN, int FOUT, bool PRE>
static void run_layer(const float* in, const float* W, const float* preb,
                      float* h, float* agg, const float* dinv,
                      const int* src, const int* dst, int n, int ne, hipStream_t s) {
  constexpr int CT = (FOUT + 15) / 16;
  constexpr int TILES_PER_WAVE = 8;
  int ntiles = (n + 15) / 16;
  int ntStride = (ntiles + TILES_PER_WAVE - 1) / TILES_PER_WAVE;
  long thr = (long)ntStride * CT * 32;
  k_gemm_wmma<FIN, FOUT, PRE><<<(int)((thr + 255) / 256), 256, 0, s>>>(
      in, W, preb, dinv, h, agg, n, ntStride);
  constexpr int CPF = (FOUT + 3) / 4;
  long te = (long)ne * CPF;
  k_scatter<FOUT><<<(int)((te + 255) / 256), 256, 0, s>>>(src, dst, dinv, h, agg, ne);
}

extern "C" void kernel_launch(void* const* d_in, const int* in_sizes, int n_in,
                              void* d_out, int out_size, void* d_ws, size_t ws_size,
                              hipStream_t stream) {
  const float* x    = (const float*)d_in[0];
  const int*  eidx  = (const int*)d_in[1];
  const int*  batch = (const int*)d_in[2];
  const float* W1 = (const float*)d_in[3];  const float* b1 = (const float*)d_in[4];
  const float* W2 = (const float*)d_in[5];  const float* b2 = (const float*)d_in[6];
  const float* W3 = (const float*)d_in[7];  const float* b3 = (const float*)d_in[8];
  const float* W4 = (const float*)d_in[9];  const float* b4 = (const float*)d_in[10];
  float* out = (float*)d_out;

  int n  = in_sizes[0] / 3;   // 100000 nodes
  int ne = in_sizes[1] / 2;   // 3200000 edges
  const int* src = eidx;
  const int* dst = eidx + ne;

  // workspace layout (256B aligned regions)
  auto al = [](size_t v) { return (v + 255) & ~(size_t)255; };
  char* ws = (char*)d_ws;
  float* dinv = (float*)ws;                                  // n floats
  size_t off  = al((size_t)n * 4);
  float* hbuf = (float*)(ws + off);                          // n*64 floats
  off += al((size_t)n * 64 * 4);
  float* agg  = (float*)(ws + off);                          // n*64 floats
  off += al((size_t)n * 64 * 4);
  float* sums = (float*)(ws + off);                          // 64*6 floats
  float* cnt  = sums + NGRAPHS * 6;                          // 64 floats (contiguous)

  // 1) degrees -> dinv
  k_deg_init<<<(n + 255) / 256, 256, 0, stream>>>(dinv, n);
  k_deg_accum<<<(ne + 255) / 256, 256, 0, stream>>>(dst, dinv, ne);
  k_deg_rsqrt<<<(n + 255) / 256, 256, 0, stream>>>(dinv, n);

  // 2) four GCN layers; bias+ReLU of layer L fused into layer L+1's GEMM A-load
  run_layer<3, 16, false>(x,   W1, x,  hbuf, agg, dinv, src, dst, n, ne, stream);
  run_layer<16, 64, true>(agg, W2, b1, hbuf, agg, dinv, src, dst, n, ne, stream);
  run_layer<64, 16, true>(agg, W3, b2, hbuf, agg, dinv, src, dst, n, ne, stream);
  run_layer<16, 6,  true>(agg, W4, b3, hbuf, agg, dinv, src, dst, n, ne, stream);

  // 3) global mean pool + bias4 + log_softmax
  k_zero<<<2, 256, 0, stream>>>(sums, NGRAPHS * 6 + NGRAPHS);
  k_pool<<<(n + 255) / 256, 256, 0, stream>>>(agg, batch, sums, cnt, n);
  k_finalize<<<1, NGRAPHS, 0, stream>>>(sums, cnt, b4, out);
}